// GCN_Encoder_19791209300728
// MI455X (gfx1250) — compile-verified
//
#include <hip/hip_runtime.h>
#include <hip/hip_bf16.h>

typedef __attribute__((ext_vector_type(2))) float v2f;
typedef __attribute__((ext_vector_type(8))) float v8f;

// D = A(16x4 f32) * B(4x16 f32) + C(16x16 f32)   -> v_wmma_f32_16x16x4_f32
__device__ __forceinline__ v8f wmma4(v2f a, v2f b, v8f c) {
  return __builtin_amdgcn_wmma_f32_16x16x4_f32(
      false, a, false, b, (short)0, c, false, false);
}

// ---------------------------------------------------------------------------
// GCN layer GEMM: H = act(X) @ W  (D fixed = 128), fused epilogue:
//   H[r,c]   = h
//   ACC[r,c] = h * deg_inv[r] + bias[c]        (self-loop + bias pre-init)
// One wave computes a 16x128 strip (8 WMMA accumulators). W staged in LDS.
// ---------------------------------------------------------------------------
template <int RELU_IN>
__global__ __launch_bounds__(256)
void gcn_gemm_kernel(const float* __restrict__ X, const float* __restrict__ W,
                     const float* __restrict__ bias, const float* __restrict__ deg_inv,
                     float* __restrict__ H, float* __restrict__ ACC,
                     int M, int K)
{
  constexpr int D = 128;
  __shared__ float wlds[128 * D];            // 64 KB, K <= 128

  // Cooperative b128 staging of W into LDS (K*D is a multiple of 4).
  {
    const int nvec = (K * D) >> 2;
    const float4* __restrict__ wv = (const float4*)W;
    float4* __restrict__ lv = (float4*)wlds;
    for (int i = threadIdx.x; i < nvec; i += 256) lv[i] = wv[i];
  }
  __syncthreads();

  const int gtid = blockIdx.x * blockDim.x + threadIdx.x;
  const int wave = gtid >> 5;
  const int lane = threadIdx.x & 31;
  const int half = lane >> 4;                // lane group 0..1
  const int l    = lane & 15;
  const int m0   = wave * 16;
  if (m0 >= M) return;                       // wave-uniform; after the barrier

  const int arow = (m0 + l < M) ? (m0 + l) : (M - 1);
  const float* __restrict__ xrow = X + (size_t)arow * K;

  v8f acc[8] = {};
  const int Kmain = K & ~3;

  // Running pointers: no address recomputation in the loop.
  const float* __restrict__ xp = xrow + 2 * half;       // A: K = kb, kb+1
  const float* __restrict__ wp = wlds + 2 * half * D + l;

  // Main loop: unguarded; aligned v2f A load; batch the 8 B-fragment LDS
  // loads before the 8 WMMAs so they issue as one clause.
#pragma unroll 2
  for (int k = 0; k < Kmain; k += 4) {
    v2f a = *(const v2f*)xp;
    if (RELU_IN) { a.x = fmaxf(a.x, 0.f); a.y = fmaxf(a.y, 0.f); }
    v2f b[8];
#pragma unroll
    for (int t = 0; t < 8; ++t) {
      b[t].x = wp[t * 16];
      b[t].y = wp[D + t * 16];
    }
#pragma unroll
    for (int t = 0; t < 8; ++t) acc[t] = wmma4(a, b[t], acc[t]);
    xp += 4;
    wp += 4 * D;
  }

  // Tail (only when K % 4 != 0, e.g. K = 78): one guarded iteration.
  if (Kmain < K) {
    const int kb = Kmain + 2 * half;
    float a0 = (kb     < K) ? xrow[kb]     : 0.f;
    float a1 = (kb + 1 < K) ? xrow[kb + 1] : 0.f;
    if (RELU_IN) { a0 = fmaxf(a0, 0.f); a1 = fmaxf(a1, 0.f); }
    v2f a; a.x = a0; a.y = a1;
    v2f b[8];
#pragma unroll
    for (int t = 0; t < 8; ++t) {
      const int col = t * 16 + l;
      b[t].x = (kb     < K) ? wlds[kb * D + col]       : 0.f;
      b[t].y = (kb + 1 < K) ? wlds[(kb + 1) * D + col] : 0.f;
    }
#pragma unroll
    for (int t = 0; t < 8; ++t) acc[t] = wmma4(a, b[t], acc[t]);
  }

  // Epilogue: C/D layout -> row = m0 + v + 8*(lane>=16), col = t*16 + (lane&15)
#pragma unroll
  for (int v = 0; v < 8; ++v) {
    const int row = m0 + v + 8 * half;
    if (row < M) {
      const float dinv = deg_inv[row];
      float* __restrict__ hrow = H + (size_t)row * D + l;
      float* __restrict__ arowp = ACC + (size_t)row * D + l;
#pragma unroll
      for (int t = 0; t < 8; ++t) {
        const float h = acc[t][v];
        hrow[t * 16]  = h;
        arowp[t * 16] = fmaf(h, dinv, bias[t * 16 + l]);
      }
    }
  }
}

// ---------------------------------------------------------------------------
// FF head GEMM: Y = relu(X @ W + bias). Wave -> 16x128; col groups for D=256.
// K is always a multiple of 4 here -> fully unguarded inner loop.
// ---------------------------------------------------------------------------
__global__ __launch_bounds__(256)
void ff_gemm_kernel(const float* __restrict__ X, const float* __restrict__ W,
                    const float* __restrict__ bias, float* __restrict__ Y,
                    int M, int K, int D)
{
  const int gtid = blockIdx.x * blockDim.x + threadIdx.x;
  const int wave = gtid >> 5;
  const int lane = threadIdx.x & 31;
  const int half = lane >> 4;
  const int l    = lane & 15;
  const int cgroups = D >> 7;                // D / 128
  const int cg = wave % cgroups;
  const int m0 = (wave / cgroups) * 16;
  const int n0 = cg * 128;
  if (m0 >= M) return;

  const int arow = (m0 + l < M) ? (m0 + l) : (M - 1);
  const float* __restrict__ xp = X + (size_t)arow * K + 2 * half;
  const float* __restrict__ wp = W + (size_t)(2 * half) * D + n0 + l;
  const size_t wstep = (size_t)4 * D;

  v8f acc[8] = {};
  for (int k = 0; k < K; k += 4) {
    v2f a = *(const v2f*)xp;
    v2f b[8];
#pragma unroll
    for (int t = 0; t < 8; ++t) {
      b[t].x = wp[t * 16];
      b[t].y = wp[D + t * 16];
    }
#pragma unroll
    for (int t = 0; t < 8; ++t) acc[t] = wmma4(a, b[t], acc[t]);
    xp += 4;
    wp += wstep;
  }
#pragma unroll
  for (int v = 0; v < 8; ++v) {
    const int row = m0 + v + 8 * half;
    if (row < M) {
      float* __restrict__ yrow = Y + (size_t)row * D + n0 + l;
#pragma unroll
      for (int t = 0; t < 8; ++t) {
        yrow[t * 16] = fmaxf(acc[t][v] + bias[n0 + t * 16 + l], 0.f);
      }
    }
  }
}

// ---------------------------------------------------------------------------
// Degree / normalization
// ---------------------------------------------------------------------------
__global__ void deg_init_kernel(float* __restrict__ deg, int N) {
  int i = blockIdx.x * blockDim.x + threadIdx.x;
  if (i < N) deg[i] = 1.0f;                  // +1 for the self-loop
}
__global__ void deg_count_kernel(const int* __restrict__ dst, float* __restrict__ deg, int E) {
  int i = blockIdx.x * blockDim.x + threadIdx.x;
  if (i < E) atomicAdd(&deg[dst[i]], 1.0f);
}
__global__ void deg_fin_kernel(const float* __restrict__ deg, float* __restrict__ dis,
                               float* __restrict__ dinv, int N) {
  int i = blockIdx.x * blockDim.x + threadIdx.x;
  if (i < N) {
    float r = rsqrtf(deg[i]);
    dis[i]  = r;
    dinv[i] = r * r;                         // matches deg_isqrt * deg_isqrt
  }
}

// ---------------------------------------------------------------------------
// Edge scatter: ACC[dst] += H[src] * (dis[src]*dis[dst]); 32 lanes per edge.
// ---------------------------------------------------------------------------
__global__ void scatter_kernel(const float* __restrict__ H, const int* __restrict__ src,
                               const int* __restrict__ dst, const float* __restrict__ dis,
                               float* __restrict__ ACC, int E)
{
  constexpr int D = 128;
  int gid = blockIdx.x * blockDim.x + threadIdx.x;
  int e = gid >> 5;
  int lane = gid & 31;
  if (e >= E) return;
  int s = src[e], d = dst[e];
  float norm = dis[s] * dis[d];
  const float4 h4 = *(const float4*)(H + (size_t)s * D + lane * 4);
  float* out = ACC + (size_t)d * D + lane * 4;
  atomicAdd(out + 0, h4.x * norm);
  atomicAdd(out + 1, h4.y * norm);
  atomicAdd(out + 2, h4.z * norm);
  atomicAdd(out + 3, h4.w * norm);
}

// ---------------------------------------------------------------------------
// Global max pool (post-ReLU values >= 0 -> int-compare atomicMax is exact;
// init 0 also implements the isfinite->0 clamp for empty graphs).
// ---------------------------------------------------------------------------
__global__ void pool_init_kernel(float* __restrict__ pooled, int n) {
  int i = blockIdx.x * blockDim.x + threadIdx.x;
  if (i < n) pooled[i] = 0.0f;
}
__global__ void pool_max_kernel(const float* __restrict__ ACC, const int* __restrict__ batch,
                                float* __restrict__ pooled, int N)
{
  constexpr int D = 128;
  int gid = blockIdx.x * blockDim.x + threadIdx.x;
  int i = gid >> 7;
  int c = gid & 127;
  if (i >= N) return;
  float v = fmaxf(ACC[(size_t)i * D + c], 0.0f);   // final-layer ReLU fused here
  int g = batch[i];
  atomicMax((int*)(pooled + (size_t)g * D + c), __float_as_int(v));
}

// ---------------------------------------------------------------------------
extern "C" void kernel_launch(void* const* d_in, const int* in_sizes, int n_in,
                              void* d_out, int out_size, void* d_ws, size_t ws_size,
                              hipStream_t stream)
{
  const float* x     = (const float*)d_in[0];
  const int*   eidx  = (const int*)d_in[1];
  const int*   batch = (const int*)d_in[2];
  // d_in[3] = n_graphs scalar (unused; derived from out_size)
  const float* gW0 = (const float*)d_in[4];
  const float* gb0 = (const float*)d_in[5];
  const float* gW1 = (const float*)d_in[6];
  const float* gb1 = (const float*)d_in[7];
  const float* gW2 = (const float*)d_in[8];
  const float* gb2 = (const float*)d_in[9];
  const float* fW0 = (const float*)d_in[10];
  const float* fb0 = (const float*)d_in[11];
  const float* fW1 = (const float*)d_in[12];
  const float* fb1 = (const float*)d_in[13];

  const int IN_DIM = 78, D = 128, FD1 = 256;
  const int N = in_sizes[0] / IN_DIM;
  const int E = in_sizes[1] / 2;
  const int G = out_size / D;
  const int* src = eidx;
  const int* dst = eidx + E;

  char* ws = (char*)d_ws;
  size_t off = 0;
  auto take = [&](size_t bytes) -> void* {
    void* p = ws + off;
    off += (bytes + 255) & ~(size_t)255;
    return p;
  };
  float* H      = (float*)take((size_t)N * D * sizeof(float));
  float* ACCa   = (float*)take((size_t)N * D * sizeof(float));
  float* ACCb   = (float*)take((size_t)N * D * sizeof(float));
  float* deg    = (float*)take((size_t)N * sizeof(float));
  float* dis    = (float*)take((size_t)N * sizeof(float));
  float* dinv   = (float*)take((size_t)N * sizeof(float));
  float* pooled = (float*)take((size_t)G * D * sizeof(float));
  float* ffh    = (float*)take((size_t)G * FD1 * sizeof(float));
  (void)ws_size; (void)n_in;

  const dim3 b256(256);
  const int nblk  = (N + 255) / 256;
  const int eblk  = (E + 255) / 256;
  const int sblk  = (E * 32 + 255) / 256;
  const int gemmb = (((N + 15) / 16) * 32 + 255) / 256;

  deg_init_kernel<<<nblk, b256, 0, stream>>>(deg, N);
  deg_count_kernel<<<eblk, b256, 0, stream>>>(dst, deg, E);
  deg_fin_kernel<<<nblk, b256, 0, stream>>>(deg, dis, dinv, N);

  gcn_gemm_kernel<0><<<gemmb, b256, 0, stream>>>(x, gW0, gb0, dinv, H, ACCa, N, IN_DIM);
  scatter_kernel<<<sblk, b256, 0, stream>>>(H, src, dst, dis, ACCa, E);

  gcn_gemm_kernel<1><<<gemmb, b256, 0, stream>>>(ACCa, gW1, gb1, dinv, H, ACCb, N, D);
  scatter_kernel<<<sblk, b256, 0, stream>>>(H, src, dst, dis, ACCb, E);

  gcn_gemm_kernel<1><<<gemmb, b256, 0, stream>>>(ACCb, gW2, gb2, dinv, H, ACCa, N, D);
  scatter_kernel<<<sblk, b256, 0, stream>>>(H, src, dst, dis, ACCa, E);

  pool_init_kernel<<<(G * D + 255) / 256, b256, 0, stream>>>(pooled, G * D);
  pool_max_kernel<<<(N * 128 + 255) / 256, b256, 0, stream>>>(ACCa, batch, pooled, N);

  {
    int waves = ((G + 15) / 16) * (FD1 / 128);
    ff_gemm_kernel<<<(waves * 32 + 255) / 256, b256, 0, stream>>>(pooled, fW0, fb0, ffh, G, D, FD1);
  }
  {
    int waves = ((G + 15) / 16) * (D / 128);
    ff_gemm_kernel<<<(waves * 32 + 255) / 256, b256, 0, stream>>>(ffh, fW1, fb1, (float*)d_out, G, FD1, D);
  }
}